// NousAlgebra_50620484551273
// MI455X (gfx1250) — compile-verified
//
#include <hip/hip_runtime.h>
#include <math.h>

// ---------------------------------------------------------------------------
// Durand-Kerner root finder for B=65536 degree-15 real-coefficient polynomials.
// Compute-bound f32 VALU problem (~32 GFLOP vs ~12MB traffic @ 23.3 TB/s).
// CDNA5 paths used:
//   * V_WMMA_F32_16X16X4_F32 rank-1 outer product for the initial guesses
//     (radius[b] x e^{i theta_j}), 16 polys x 16 roots per wave, f32-exact.
//   * GLOBAL_LOAD_ASYNC_TO_LDS_* to stage each thread's root state into LDS.
// Solver: 1 thread = 1 polynomial; roots double-buffered in LDS (dynamic j/k
// indexing stays in LDS, coefficients stay in VGPRs via unrolled Horner).
// ---------------------------------------------------------------------------

#define GLOBAL_AS __attribute__((address_space(1)))
#define LOCAL_AS  __attribute__((address_space(3)))

typedef __attribute__((ext_vector_type(2))) float v2f;
typedef __attribute__((ext_vector_type(8))) float v8f;
typedef __attribute__((ext_vector_type(4))) int   v4i;
typedef __attribute__((ext_vector_type(2))) int   v2i;

#define NP1 16
#define NR  15
#define DK_ITERS     60
#define HALLEY_ITERS 30
#define NEWTON_ITERS 20
#define THR2 1e-12f               /* (1e-6)^2 cluster threshold, squared */
#define TWO_PI_OVER_N 0.4188790204786391f
#define TINY2 1e-37f              /* "effectively zero" |.|^2 guard */

// Deterministic small per-root perturbation (reference jitters angles with a
// JAX PRNG we cannot reproduce; magnitude-matched 0.01 jitter breaks symmetry
// ties identically in effect).
__device__ __forceinline__ float root_pert_re(int j) { return 0.01f * __sinf(1.7f * (float)j + 0.3f); }
__device__ __forceinline__ float root_pert_im(int j) { return 0.01f * __cosf(2.3f * (float)j + 0.7f); }

// ---------------------------------------------------------------------------
// Kernel 1: initial guesses via f32 WMMA outer product.
//   D[poly, root] = radius[poly] * e[root]  (K=1, zero-padded to 16x16x4)
// A layout (16x4 f32, documented): lanes 0-15 M=0-15 {VGPR0=K0, VGPR1=K1},
//   lanes 16-31 M=0-15 {VGPR0=K2, VGPR1=K3}  -> radius in lanes<16 comp .x.
// B layout (4x16 f32, mirror of A):  lanes 0-15 N=0-15 {VGPR0=K0,...}
//   -> unit-circle nodes in lanes<16 comp .x, everything else zero.
// D layout (16x16 f32, documented): VGPR r: lanes0-15 -> M=r,N=lane;
//   lanes16-31 -> M=r+8, N=lane-16.
// ---------------------------------------------------------------------------
__global__ __launch_bounds__(256) void dk_init(const float* __restrict__ coeffs,
                                               float* __restrict__ z0,
                                               int npoly)
{
    const int lane     = threadIdx.x & 31;
    const int wave     = blockIdx.x * (blockDim.x >> 5) + (threadIdx.x >> 5);
    const int polyBase = wave * 16;
    if (polyBase >= npoly) return;                 // wave-uniform

    const bool fullTile = (polyBase + 16) <= npoly;

#if __has_builtin(__builtin_amdgcn_wmma_f32_16x16x4_f32)
    if (fullTile) {
        // radius = 0.5 + 0.6 * |c1/c0|^(1/15)   (row value, lanes hold M=lane&15)
        float radius;
        {
            const int p  = polyBase + (lane & 15);
            const float c0 = coeffs[(size_t)p * NP1 + 0];
            const float c1 = coeffs[(size_t)p * NP1 + 1];
            radius = 0.5f + 0.6f * __powf(fabsf(c1 / c0), 1.0f / 15.0f);
        }
        // unit-circle nodes (column value); column 15 unused -> 0
        float er = 0.0f, ei = 0.0f;
        {
            const int j = lane & 15;
            if (j < NR) {
                const float th = (float)j * TWO_PI_OVER_N;
                __sincosf(th, &ei, &er);
            }
        }
        v2f a;  a.x  = (lane < 16) ? radius : 0.0f;  a.y  = 0.0f;  // only K=0 row
        v2f br; br.x = (lane < 16) ? er     : 0.0f;  br.y = 0.0f;
        v2f bi; bi.x = (lane < 16) ? ei     : 0.0f;  bi.y = 0.0f;
        v8f c = {};
        v8f dre = __builtin_amdgcn_wmma_f32_16x16x4_f32(false, a, false, br, (short)0, c, false, false);
        v8f dim = __builtin_amdgcn_wmma_f32_16x16x4_f32(false, a, false, bi, (short)0, c, false, false);

        const int root = lane & 15;
        const int half = (lane >> 4) * 8;
        if (root < NR) {
            const float pre = root_pert_re(root);
            const float pim = root_pert_im(root);
#pragma unroll
            for (int r = 0; r < 8; ++r) {
                const int poly = polyBase + half + r;
                z0[(size_t)poly * NR * 2 + root * 2 + 0] = dre[r] + pre;
                z0[(size_t)poly * NR * 2 + root * 2 + 1] = dim[r] + pim;
            }
        }
        return;
    }
#endif
    // Scalar fallback: partial tile (or builtin unavailable).
    if (lane < 16) {
        const int p = polyBase + lane;
        if (p < npoly) {
            const float c0 = coeffs[(size_t)p * NP1 + 0];
            const float c1 = coeffs[(size_t)p * NP1 + 1];
            const float radius = 0.5f + 0.6f * __powf(fabsf(c1 / c0), 1.0f / 15.0f);
            for (int j = 0; j < NR; ++j) {
                float s, cx;
                __sincosf((float)j * TWO_PI_OVER_N, &s, &cx);
                z0[(size_t)p * NR * 2 + j * 2 + 0] = radius * cx + root_pert_re(j);
                z0[(size_t)p * NR * 2 + j * 2 + 1] = radius * s  + root_pert_im(j);
            }
        }
    }
}

// ---------------------------------------------------------------------------
// Kernel 2: the full solver. One thread per polynomial. Roots live in LDS
// (double-buffered, Jacobi-style updates as in the reference). zio holds the
// WMMA-produced initial guesses on entry and the final roots [b][15][2] on
// exit.
// ---------------------------------------------------------------------------
__global__ __launch_bounds__(256) void dk_solve(const float* __restrict__ coeffs,
                                                float* __restrict__ zio,
                                                int npoly)
{
    const int t = blockIdx.x * 256 + threadIdx.x;
    __shared__ float zs0[256 * NR * 2];   // 30 KB
    __shared__ float zs1[256 * NR * 2];   // 30 KB
    if (t >= npoly) return;

    // --- normalized coefficients in registers (leading coeff -> 1) ---
    float a[NP1];
    {
        const float4* cp = (const float4*)(coeffs + (size_t)t * NP1);
        const float4 q0 = cp[0], q1 = cp[1], q2 = cp[2], q3 = cp[3];
        const float inv = 1.0f / q0.x;
        a[0]  = 1.0f;       a[1]  = q0.y * inv; a[2]  = q0.z * inv; a[3]  = q0.w * inv;
        a[4]  = q1.x * inv; a[5]  = q1.y * inv; a[6]  = q1.z * inv; a[7]  = q1.w * inv;
        a[8]  = q2.x * inv; a[9]  = q2.y * inv; a[10] = q2.z * inv; a[11] = q2.w * inv;
        a[12] = q3.x * inv; a[13] = q3.y * inv; a[14] = q3.z * inv; a[15] = q3.w * inv;
    }

    float* zc = &zs0[threadIdx.x * NR * 2];
    float* zn = &zs1[threadIdx.x * NR * 2];
    const float* gz = zio + (size_t)t * NR * 2;

    // --- stage initial roots global -> LDS via CDNA5 async path ---
    // Builtin signature (probe-confirmed via diagnostics): arg0 = AS1 pointer
    // to v4i/v2i (global source), arg1 = LDS destination, then imm offset,
    // imm cpol.
#if __has_builtin(__builtin_amdgcn_global_load_async_to_lds_b128) && \
    __has_builtin(__builtin_amdgcn_global_load_async_to_lds_b64)
    {
#pragma unroll
        for (int q = 0; q < 7; ++q)   // 7 x 16B
            __builtin_amdgcn_global_load_async_to_lds_b128(
                (GLOBAL_AS v4i*)(gz + q * 4),
                (LOCAL_AS v4i*)(zc + q * 4), 0, 0);
        __builtin_amdgcn_global_load_async_to_lds_b64(    // last 8B (120B total)
            (GLOBAL_AS v2i*)(gz + 28),
            (LOCAL_AS v2i*)(zc + 28), 0, 0);
#if __has_builtin(__builtin_amdgcn_s_wait_asynccnt)
        __builtin_amdgcn_s_wait_asynccnt(0);
#else
        asm volatile("s_wait_asynccnt 0" ::: "memory");
#endif
        asm volatile("" ::: "memory");
    }
#else
#pragma unroll
    for (int j = 0; j < NR; ++j) {
        zc[2 * j]     = gz[2 * j];
        zc[2 * j + 1] = gz[2 * j + 1];
    }
#endif

    // =============== Durand-Kerner: z <- z - p(z)/prod(z_j - z_k) ===========
    for (int it = 0; it < DK_ITERS; ++it) {
        for (int j = 0; j < NR; ++j) {
            const float zr = zc[2 * j], zi = zc[2 * j + 1];
            // Horner p(z) with real coefficients, a[0]=1
            float pr = 1.0f, pi = 0.0f;
#pragma unroll
            for (int k = 1; k < NP1; ++k) {
                const float tr = pr * zr - pi * zi + a[k];
                const float ti = pr * zi + pi * zr;
                pr = tr; pi = ti;
            }
            // prod over k!=j of (z_j - z_k); k==j contributes (1,0) like eye
            float dr = 1.0f, di = 0.0f;
#pragma unroll
            for (int k = 0; k < NR; ++k) {
                const float wr = (k == j) ? 1.0f : (zr - zc[2 * k]);
                const float wi = (k == j) ? 0.0f : (zi - zc[2 * k + 1]);
                const float tr = dr * wr - di * wi;
                const float ti = dr * wi + di * wr;
                dr = tr; di = ti;
            }
            dr += 1e-30f;                       // denom + 1e-30 (real part)
            const float den = dr * dr + di * di;
            const float qr = (pr * dr + pi * di) / den;
            const float qi = (pi * dr - pr * di) / den;
            zn[2 * j]     = zr - qr;
            zn[2 * j + 1] = zi - qi;
        }
        float* tp = zc; zc = zn; zn = tp;
    }

    // =============== Halley-style multiplicity-aware polish =================
    for (int it = 0; it < HALLEY_ITERS; ++it) {
        for (int j = 0; j < NR; ++j) {
            const float zr = zc[2 * j], zi = zc[2 * j + 1];
            float pr = 0.f, pi = 0.f, dpr = 0.f, dpi = 0.f, sr = 0.f, si = 0.f; // s = d2p
#pragma unroll
            for (int k = 0; k < NP1; ++k) {
                const float t2r = sr * zr - si * zi + 2.0f * dpr;
                const float t2i = sr * zi + si * zr + 2.0f * dpi;
                const float t1r = dpr * zr - dpi * zi + pr;
                const float t1i = dpr * zi + dpi * zr + pi;
                const float t0r = pr * zr - pi * zi + a[k];
                const float t0i = pr * zi + pi * zr;
                sr = t2r; si = t2i; dpr = t1r; dpi = t1i; pr = t0r; pi = t0i;
            }
            const float numr = pr * sr - pi * si;
            const float numi = pr * si + pi * sr;
            const float denr = (dpr * dpr - dpi * dpi) - numr;
            const float deni = (2.0f * dpr * dpi) - numi;
            const float dmag = denr * denr + deni * deni;
            float m = 1.0f;
            if (dmag > 0.0f)                     // |den| == 0 -> m = 1
                m = (numr * denr + numi * deni) / dmag + 1.0f;  // Re(num/den)+1
            m = fminf(fmaxf(m, 1.0f), 15.0f);
            float dpm = dpr * dpr + dpi * dpi;
            float sdr = dpr, sdi = dpi;
            if (dpm < TINY2) { sdr = 1e-30f; sdi = 0.0f; dpm = TINY2; }
            const float qr = m * (pr * sdr + pi * sdi) / dpm;
            const float qi = m * (pi * sdr - pr * sdi) / dpm;
            zn[2 * j]     = zr - qr;
            zn[2 * j + 1] = zi - qi;
        }
        float* tp = zc; zc = zn; zn = tp;
    }

    // =============== cluster collapse (diagonal-inclusive count) ============
    {
        for (int j = 0; j < NR; ++j) {
            const float zr = zc[2 * j], zi = zc[2 * j + 1];
            float cnt = 0.f, sre = 0.f, sim = 0.f;
#pragma unroll
            for (int k = 0; k < NR; ++k) {
                const float wr = zr - zc[2 * k];
                const float wi = zi - zc[2 * k + 1];
                const float d2 = wr * wr + wi * wi;
                if (d2 < THR2) { cnt += 1.0f; sre += zc[2 * k]; sim += zc[2 * k + 1]; }
            }
            float orr, oii;
            if (cnt > 14.5f)      { orr = -a[1] / 15.0f; oii = 0.0f; }   // all equal
            else if (cnt > 1.5f)  { orr = sre / cnt;     oii = sim / cnt; }
            else                  { orr = zr;            oii = zi; }
            zn[2 * j]     = orr;
            zn[2 * j + 1] = oii;
        }
        float* tp = zc; zc = zn; zn = tp;
    }

    // =============== modified Newton with multiplicity 2 ====================
    for (int it = 0; it < NEWTON_ITERS; ++it) {
        for (int j = 0; j < NR; ++j) {
            const float zr = zc[2 * j], zi = zc[2 * j + 1];
            float pr = 1.0f, pi = 0.0f, dpr = 0.0f, dpi = 0.0f;
#pragma unroll
            for (int k = 1; k < NP1; ++k) {
                const float t1r = dpr * zr - dpi * zi + pr;
                const float t1i = dpr * zi + dpi * zr + pi;
                const float t0r = pr * zr - pi * zi + a[k];
                const float t0i = pr * zi + pi * zr;
                dpr = t1r; dpi = t1i; pr = t0r; pi = t0i;
            }
            float mind2 = 3.4e38f;
#pragma unroll
            for (int k = 0; k < NR; ++k) {
                const float wr = zr - zc[2 * k];
                const float wi = zi - zc[2 * k + 1];
                float d2 = wr * wr + wi * wi;
                d2 = (k == j) ? 3.4e38f : d2;            // eye -> inf
                mind2 = fminf(mind2, d2);
            }
            const float mult = (mind2 < THR2) ? 2.0f : 1.0f;
            float dpm = dpr * dpr + dpi * dpi;
            float sdr = dpr, sdi = dpi;
            if (dpm < TINY2) { sdr = 1e-30f; sdi = 0.0f; dpm = TINY2; }
            const float qr = mult * (pr * sdr + pi * sdi) / dpm;
            const float qi = mult * (pi * sdr - pr * sdi) / dpm;
            zn[2 * j]     = zr - qr;
            zn[2 * j + 1] = zi - qi;
        }
        float* tp = zc; zc = zn; zn = tp;
    }

    // --- final store [b][15][2] ---
    float* zo = zio + (size_t)t * NR * 2;
#pragma unroll
    for (int j = 0; j < NR; ++j) {
        zo[2 * j]     = zc[2 * j];
        zo[2 * j + 1] = zc[2 * j + 1];
    }
}

// ---------------------------------------------------------------------------
extern "C" void kernel_launch(void* const* d_in, const int* in_sizes, int n_in,
                              void* d_out, int out_size, void* d_ws, size_t ws_size,
                              hipStream_t stream)
{
    (void)n_in; (void)out_size; (void)d_ws; (void)ws_size;
    const float* coeffs = (const float*)d_in[0];
    float* out = (float*)d_out;                 // [b][15][2] f32; also z0 staging
    const int npoly = in_sizes[0] / NP1;

    const int waves      = (npoly + 15) / 16;
    const int initBlocks = (waves + 7) / 8;     // 8 waves per 256-thread block
    dk_init<<<initBlocks, 256, 0, stream>>>(coeffs, out, npoly);

    const int solveBlocks = (npoly + 255) / 256;
    dk_solve<<<solveBlocks, 256, 0, stream>>>(coeffs, out, npoly);
}